// Classifier_55413668053119
// MI455X (gfx1250) — compile-verified
//
#include <hip/hip_runtime.h>
#include <hip/hip_bf16.h>

#define N_NODES   50000
#define PAD_NODES 50048         // padded row count for agg2 (multiple of 32)
#define N_EDGES   800000
#define N_GRAPHS  16
#define HIDDEN    256
#define N_CLASSES 10

typedef __attribute__((ext_vector_type(2)))  float  v2f;
typedef __attribute__((ext_vector_type(8)))  float  v8f;

// ---- workspace layout (float offsets, 64-float aligned) ----
#define OFF_INDEG  0
#define OFF_OUTDEG 50048
#define OFF_AGG1   100096
#define OFF_S      150144
#define OFF_COUNTS 200192
#define OFF_HG     200256
#define OFF_AGG2   204352
#define WS_FLOATS  (OFF_AGG2 + PAD_NODES * HIDDEN)   // 13,016,640 (mult of 4)

// -------------------- zero the workspace --------------------
__global__ void k_zero(float4* p, int n4) {
    int i = blockIdx.x * blockDim.x + threadIdx.x;
    int stride = gridDim.x * blockDim.x;
    float4 z = make_float4(0.f, 0.f, 0.f, 0.f);
    for (; i < n4; i += stride) p[i] = z;
}

// -------------------- degree accumulation --------------------
__global__ void k_degrees(const int* __restrict__ src, const int* __restrict__ dst,
                          float* in_deg, float* out_deg) {
    int i = blockIdx.x * blockDim.x + threadIdx.x;
    int stride = gridDim.x * blockDim.x;
    for (; i < N_EDGES; i += stride) {
        atomicAdd(&in_deg[dst[i]], 1.0f);
        atomicAdd(&out_deg[src[i]], 1.0f);
    }
}

// ---- layer-1 scalar aggregation: agg1[dst] += in_deg[src]*norm_src[src] ----
__global__ void k_agg1(const int* __restrict__ src, const int* __restrict__ dst,
                       const float* __restrict__ in_deg, const float* __restrict__ out_deg,
                       float* agg1) {
    int i = blockIdx.x * blockDim.x + threadIdx.x;
    int stride = gridDim.x * blockDim.x;
    for (; i < N_EDGES; i += stride) {
        int sn = src[i];
        float ns = rsqrtf(fmaxf(out_deg[sn], 1.0f));
        atomicAdd(&agg1[dst[i]], in_deg[sn] * ns);
    }
}

// ---- per-node scalar s[n] = agg1[n]*norm_dst[n]; graph counts ----
__global__ void k_s_counts(const float* __restrict__ agg1, const float* __restrict__ in_deg,
                           const int* __restrict__ gids, float* s, float* counts) {
    int n = blockIdx.x * blockDim.x + threadIdx.x;
    if (n >= N_NODES) return;
    s[n] = agg1[n] * rsqrtf(fmaxf(in_deg[n], 1.0f));
    atomicAdd(&counts[gids[n]], 1.0f);
}

// ---- layer-2 edge pass: agg2[dst] += relu(s[src]*W1 + b1) * norm_src[src] ----
// One wave per edge; lane j covers channels {j, j+32, ..., j+224}.
// h1 is never materialized: 4-byte gather of s instead of 1KB gather of h1.
__global__ __launch_bounds__(256) void k_edge_layer2(
    const int* __restrict__ src, const int* __restrict__ dst,
    const float* __restrict__ s, const float* __restrict__ out_deg,
    const float* __restrict__ W1, const float* __restrict__ b1,
    float* __restrict__ agg2) {
    const int lane = threadIdx.x & 31;
    const int wave = threadIdx.x >> 5;
    float w1r[8], b1r[8];
#pragma unroll
    for (int j = 0; j < 8; ++j) { w1r[j] = W1[lane + 32 * j]; b1r[j] = b1[lane + 32 * j]; }
    const int estride = gridDim.x * 8;
    for (int e = blockIdx.x * 8 + wave; e < N_EDGES; e += estride) {
        const int sn = src[e];
        const int dn = dst[e];
        const float sv = s[sn];
        const float ns = rsqrtf(fmaxf(out_deg[sn], 1.0f));
        float* row = agg2 + (size_t)dn * HIDDEN;
#pragma unroll
        for (int j = 0; j < 8; ++j) {
            float v = fmaxf(fmaf(sv, w1r[j], b1r[j]), 0.0f) * ns;
            atomicAdd(&row[lane + 32 * j], v);   // 32 lanes -> one 128B line
        }
    }
}

// ---- layer-2 GEMM + fused bias/ReLU/graph-mean-pool accumulation ----
// Block: 256 threads (8 waves). blockIdx.x = 32-node strip (1563 strips over the
// zero-padded 50016-row range). Each wave owns one 16-wide column tile and computes
// TWO 16x16 output tiles, sharing each B fragment across both (halves W2 traffic,
// 4 independent WMMA accumulator chains). blockIdx.y picks which 8 of 16 col tiles.
#define LDS_PITCH 260   // 256 + 4 pad: kills the 16-way bank conflict (260 % 64 == 4)
__global__ __launch_bounds__(256) void k_gemm_pool(
    const float* __restrict__ agg2, const float* __restrict__ in_deg,
    const float* __restrict__ W2, const float* __restrict__ b2,
    const int* __restrict__ gids, float* __restrict__ hg) {
    __shared__ float ldsA[32 * LDS_PITCH];
    const int tid = threadIdx.x;
    const int rowBase = blockIdx.x * 32;

    // Stage A strip (pre-scaled by norm_dst) into LDS: 32 rows x 256 f32.
    {
        const int r  = tid >> 3;           // 0..31 row
        const int c0 = (tid & 7) * 32;     // 32 floats per thread
        const int node = rowBase + r;      // < PAD_NODES, padded region is zeroed
        const float nd = rsqrtf(fmaxf(in_deg[node], 1.0f));
        const float4* g4 = (const float4*)(agg2 + (size_t)node * HIDDEN + c0);
        float* drow = ldsA + r * LDS_PITCH + c0;
#pragma unroll
        for (int q = 0; q < 8; ++q) {
            float4 v = g4[q];
            drow[q * 4 + 0] = v.x * nd;
            drow[q * 4 + 1] = v.y * nd;
            drow[q * 4 + 2] = v.z * nd;
            drow[q * 4 + 3] = v.w * nd;
        }
    }
    __syncthreads();

    const int lane  = tid & 31;
    const int wv    = tid >> 5;
    const int nTile = blockIdx.y * 8 + wv;       // 0..15
    const int n0    = nTile * 16 + (lane & 15);  // output channel
    const int hi    = lane >> 4;                 // 0 or 1 (lane half)
    const float* a0 = ldsA + (lane & 15) * LDS_PITCH;   // tile-0 row
    const float* a1 = a0 + 16 * LDS_PITCH;              // tile-1 row

    // V_WMMA_F32_16X16X4_F32: A 16x4 (lane=M; VGPR0/1 = K even/odd, +2 for hi half)
    v8f c00 = {}, c01 = {}, c10 = {}, c11 = {};
    const int krow = hi * 2;
    for (int k0 = 0; k0 < HIDDEN; k0 += 8) {
        const int kA = k0 + krow;
        const int kB = k0 + 4 + krow;
        v2f bA; bA.x = W2[kA * HIDDEN + n0]; bA.y = W2[(kA + 1) * HIDDEN + n0];
        v2f bB; bB.x = W2[kB * HIDDEN + n0]; bB.y = W2[(kB + 1) * HIDDEN + n0];
        v2f a0A = *(const v2f*)(a0 + kA);
        v2f a0B = *(const v2f*)(a0 + kB);
        v2f a1A = *(const v2f*)(a1 + kA);
        v2f a1B = *(const v2f*)(a1 + kB);
        c00 = __builtin_amdgcn_wmma_f32_16x16x4_f32(false, a0A, false, bA, (short)0, c00, false, false);
        c10 = __builtin_amdgcn_wmma_f32_16x16x4_f32(false, a1A, false, bA, (short)0, c10, false, false);
        c01 = __builtin_amdgcn_wmma_f32_16x16x4_f32(false, a0B, false, bB, (short)0, c01, false, false);
        c11 = __builtin_amdgcn_wmma_f32_16x16x4_f32(false, a1B, false, bB, (short)0, c11, false, false);
    }

    // Epilogue: bias + ReLU + mean-pool accumulation (h2 never hits memory).
    // C/D layout: VGPR i -> row M = i + 8*hi, col N = lane&15.
    // graph_ids is sorted, so a lane's 8 rows almost always share one graph:
    // reduce in-register and emit a single atomic (8x fewer hg atomics).
    const float bias = b2[n0];
#pragma unroll
    for (int t = 0; t < 2; ++t) {
        const v8f cc = (t == 0) ? (c00 + c01) : (c10 + c11);
        const int nodeBase = rowBase + t * 16 + hi * 8;
        float vals[8];
        int   gs[8];
        bool  uni = true;
#pragma unroll
        for (int i = 0; i < 8; ++i) {
            const int node = nodeBase + i;
            vals[i] = fmaxf(cc[i] + bias, 0.0f);
            gs[i]   = (node < N_NODES) ? gids[node] : -1;
            if (gs[i] != gs[0]) uni = false;
        }
        if (uni) {
            if (gs[0] >= 0) {
                float sum = 0.0f;
#pragma unroll
                for (int i = 0; i < 8; ++i) sum += vals[i];
                atomicAdd(&hg[gs[0] * HIDDEN + n0], sum);
            }
        } else {
#pragma unroll
            for (int i = 0; i < 8; ++i)
                if (gs[i] >= 0) atomicAdd(&hg[gs[i] * HIDDEN + n0], vals[i]);
        }
    }
}

// -------------------- tiny head: (hg/counts) @ Wc + bc --------------------
__global__ void k_final(const float* __restrict__ hg, const float* __restrict__ counts,
                        const float* __restrict__ Wc, const float* __restrict__ bc,
                        float* __restrict__ out) {
    const int t = threadIdx.x;
    if (t >= N_GRAPHS * N_CLASSES) return;
    const int g = t / N_CLASSES, c = t % N_CLASSES;
    const float inv = 1.0f / fmaxf(counts[g], 1.0f);
    float acc = bc[c];
    for (int k = 0; k < HIDDEN; ++k)
        acc = fmaf(hg[g * HIDDEN + k] * inv, Wc[k * N_CLASSES + c], acc);
    out[t] = acc;
}

extern "C" void kernel_launch(void* const* d_in, const int* in_sizes, int n_in,
                              void* d_out, int out_size, void* d_ws, size_t ws_size,
                              hipStream_t stream) {
    const int*   src  = (const int*)d_in[0];
    const int*   dst  = (const int*)d_in[1];
    const int*   gids = (const int*)d_in[2];
    const float* W1   = (const float*)d_in[3];
    const float* b1   = (const float*)d_in[4];
    const float* W2   = (const float*)d_in[5];
    const float* b2   = (const float*)d_in[6];
    const float* Wc   = (const float*)d_in[7];
    const float* bc   = (const float*)d_in[8];

    float* ws      = (float*)d_ws;
    float* in_deg  = ws + OFF_INDEG;
    float* out_deg = ws + OFF_OUTDEG;
    float* agg1    = ws + OFF_AGG1;
    float* sbuf    = ws + OFF_S;
    float* counts  = ws + OFF_COUNTS;
    float* hg      = ws + OFF_HG;
    float* agg2    = ws + OFF_AGG2;

    k_zero<<<4096, 256, 0, stream>>>((float4*)d_ws, WS_FLOATS / 4);
    k_degrees<<<1024, 256, 0, stream>>>(src, dst, in_deg, out_deg);
    k_agg1<<<1024, 256, 0, stream>>>(src, dst, in_deg, out_deg, agg1);
    k_s_counts<<<(N_NODES + 255) / 256, 256, 0, stream>>>(agg1, in_deg, gids, sbuf, counts);
    k_edge_layer2<<<4096, 256, 0, stream>>>(src, dst, sbuf, out_deg, W1, b1, agg2);
    dim3 gg(1563, 2);   // 1563*32 = 50016 rows (agg2 zero-padded to 50048)
    k_gemm_pool<<<gg, 256, 0, stream>>>(agg2, in_deg, W2, b2, gids, hg);
    k_final<<<1, 256, 0, stream>>>(hg, counts, Wc, bc, (float*)d_out);
}